// myLSTM_20675972563001
// MI455X (gfx1250) — compile-verified
//
#include <hip/hip_runtime.h>
#include <math.h>

typedef float v2f __attribute__((ext_vector_type(2)));
typedef float v8f __attribute__((ext_vector_type(8)));

// ---------------------------------------------------------------------------
// fp32 WMMA GEMM:  C[M,N] = A[M,K] @ W[N,K]^T + bias[N]
// One wave -> 16(M) x 64(N) strip, 4 accumulators, K stepped by 4
// (V_WMMA_F32_16X16X4_F32).  K,N compile-time -> immediate-offset loads.
// Fragment layouts per CDNA5 ISA 7.12.2:
//   A 16x4 : lanes 0-15 hold M=lane (K=k,k+1 in v0,v1); lanes 16-31 K=k+2,k+3
//   B 4x16 : lane&15 = N column, lane>>4 selects the K pair
//   C/D    : VGPR r -> M = r + 8*(lane>>4), N = lane&15
// ---------------------------------------------------------------------------
template<int K, int N>
__global__ void __launch_bounds__(256)
wmma_gemm_f32(const float* __restrict__ A,
              const float* __restrict__ W,
              const float* __restrict__ bias,
              float* __restrict__ C,
              int M)
{
    const int lane = threadIdx.x & 31;
    const int waveInBlock = threadIdx.x >> 5;
    const int gw = blockIdx.x * (blockDim.x >> 5) + waveInBlock;

    constexpr int tilesN = N >> 6;             // strips of 64 in N
    const int tileM = (gw / tilesN) << 4;
    const int tileN = (gw % tilesN) << 6;
    if (tileM >= M) return;

    const int half = lane >> 4;                // 0 | 1 -> K pair select
    const int sub  = lane & 15;                // M row (A) / N col (B)

    const float* __restrict__ arow = A + (size_t)(tileM + sub) * K + 2 * half;
    const float* __restrict__ w0 = W + (size_t)(tileN +  0 + sub) * K + 2 * half;
    const float* __restrict__ w1 = W + (size_t)(tileN + 16 + sub) * K + 2 * half;
    const float* __restrict__ w2 = W + (size_t)(tileN + 32 + sub) * K + 2 * half;
    const float* __restrict__ w3 = W + (size_t)(tileN + 48 + sub) * K + 2 * half;

    // one prefetch per source row (whole-row working set is tiny; rows are
    // L2-resident and re-read by many waves -> WGP-scope hint)
    __builtin_prefetch(arow, 0, 3);
    __builtin_prefetch(w0, 0, 3);
    __builtin_prefetch(w1, 0, 3);
    __builtin_prefetch(w2, 0, 3);
    __builtin_prefetch(w3, 0, 3);

    v8f acc0 = {}, acc1 = {}, acc2 = {}, acc3 = {};

    #pragma unroll 8
    for (int k = 0; k < K; k += 4) {
        v2f a  = *(const v2f*)(arow + k);
        v2f b0 = *(const v2f*)(w0 + k);
        v2f b1 = *(const v2f*)(w1 + k);
        v2f b2 = *(const v2f*)(w2 + k);
        v2f b3 = *(const v2f*)(w3 + k);

        acc0 = __builtin_amdgcn_wmma_f32_16x16x4_f32(false, a, false, b0,
                                                     (short)0, acc0, false, false);
        acc1 = __builtin_amdgcn_wmma_f32_16x16x4_f32(false, a, false, b1,
                                                     (short)0, acc1, false, false);
        acc2 = __builtin_amdgcn_wmma_f32_16x16x4_f32(false, a, false, b2,
                                                     (short)0, acc2, false, false);
        acc3 = __builtin_amdgcn_wmma_f32_16x16x4_f32(false, a, false, b3,
                                                     (short)0, acc3, false, false);
    }

    const float bv0 = bias[tileN +  0 + sub];
    const float bv1 = bias[tileN + 16 + sub];
    const float bv2 = bias[tileN + 32 + sub];
    const float bv3 = bias[tileN + 48 + sub];

    #pragma unroll
    for (int r = 0; r < 8; ++r) {
        const int m = tileM + r + 8 * half;
        float* crow = C + (size_t)m * N + tileN;
        crow[ 0 + sub] = acc0[r] + bv0;
        crow[16 + sub] = acc1[r] + bv1;
        crow[32 + sub] = acc2[r] + bv2;
        crow[48 + sub] = acc3[r] + bv3;
    }
}

// ---------------------------------------------------------------------------
// partial dot of W-row segment against LDS vector segment
// ---------------------------------------------------------------------------
__device__ __forceinline__ float dotseg(const float* __restrict__ w,
                                        const float* __restrict__ v,
                                        int base, int len)
{
    float s = 0.f;
    #pragma unroll 4
    for (int k = 0; k < len; k += 4) {
        float4 a = *(const float4*)(w + base + k);
        s += a.x * v[base + k]     + a.y * v[base + k + 1]
           + a.z * v[base + k + 2] + a.w * v[base + k + 3];
    }
    return s;
}

// ---------------------------------------------------------------------------
// Sequential LSTM scan for batch element 63 (the only one reaching d_out).
// Single workgroup, 1024 threads = 4 gates x 256 units; h,c in LDS.
// Work-balanced: i/f peephole dots split across gate(0,2)/(1,3) partners,
// o-peephole split 4 ways.  Reference quirk preserved:
//   h_new = sigmoid(o_pre) * tanh(c_new)^2
// ---------------------------------------------------------------------------
__global__ void __launch_bounds__(1024)
lstm_scan(const float* __restrict__ gx,   // T x 1024  (already x@Wx^T + bx)
          const float* __restrict__ Wh,   // 1024 x 256
          const float* __restrict__ bh,   // 1024
          const float* __restrict__ Wc,   // 768 x 256  (Wic|Wfc|Woc)
          const float* __restrict__ bc,   // 768
          float* __restrict__ hidden,     // T x 256
          int T)
{
    __shared__ float h[256], c[256];
    __shared__ float gi[256], gf[256], gcand[256], gopre[256], tcs[256];
    __shared__ float ppI[256], ppF[256];     // second-half peephole partials
    __shared__ float ppO[4][256];            // o-peephole quarter partials

    const int tid  = threadIdx.x;
    const int gate = tid >> 8;      // 0=i 1=f 2=cand 3=o
    const int unit = tid & 255;

    if (tid < 256) { h[tid] = 0.f; c[tid] = 0.f; }
    __syncthreads();

    const float* __restrict__ whrow = Wh + (size_t)tid * 256;
    const float  bhv = bh[tid];

    const float* __restrict__ wic = Wc + (size_t)unit * 256;          // gates 0,2
    const float* __restrict__ wfc = Wc + (size_t)(256 + unit) * 256;  // gates 1,3
    const float* __restrict__ woc = Wc + (size_t)(512 + unit) * 256;  // all, quarter
    const float bic = bc[unit];
    const float bfc = bc[256 + unit];
    const float boc = bc[512 + unit];

    for (int t = 0; t < T; ++t) {
        // ---- P1: Wh matvec + split i/f peephole halves ----
        float acc = gx[(size_t)t * 1024 + tid] + bhv + dotseg(whrow, h, 0, 256);
        float pp = 0.f;
        if (gate == 0) {
            pp = dotseg(wic, c, 0, 128);          // keep in reg
        } else if (gate == 1) {
            pp = dotseg(wfc, c, 0, 128);          // keep in reg
        } else if (gate == 2) {
            ppI[unit]   = dotseg(wic, c, 128, 128);
            gcand[unit] = tanhf(acc);
        } else {
            ppF[unit]   = dotseg(wfc, c, 128, 128);
            gopre[unit] = acc;
        }
        __syncthreads();

        // ---- P2: finalize input/forget gates ----
        if (gate == 0) {
            gi[unit] = 1.f / (1.f + expf(-(acc + pp + ppI[unit] + bic)));
        } else if (gate == 1) {
            gf[unit] = 1.f / (1.f + expf(-(acc + pp + ppF[unit] + bfc)));
        }
        __syncthreads();

        // ---- P3: cell update ----
        if (gate == 0) {
            float cn = gf[unit] * c[unit] + gi[unit] * gcand[unit];
            c[unit]   = cn;
            tcs[unit] = tanhf(cn);
        }
        __syncthreads();

        // ---- P4: o-peephole on c_new, quarter per gate ----
        ppO[gate][unit] = dotseg(woc, c, gate * 64, 64);
        __syncthreads();

        // ---- P5: output gate + h write ----
        if (gate == 3) {
            float p  = ppO[0][unit] + ppO[1][unit] + ppO[2][unit] + ppO[3][unit];
            float tc = tcs[unit];
            float o  = 1.f / (1.f + expf(-(gopre[unit] + p + boc))) * tc; // ref: o=sig(..)*tc
            float hn = o * tc;                                            // ref: h=o*tc
            h[unit] = hn;
            hidden[(size_t)t * 256 + unit] = hn;
        }
        __syncthreads();
    }
}

// ---------------------------------------------------------------------------
// In-place row-wise log_softmax over rows of length N (one block per row).
// ---------------------------------------------------------------------------
__global__ void __launch_bounds__(256)
logsoftmax_rows(float* __restrict__ data, int N)
{
    float* p = data + (size_t)blockIdx.x * N;
    __shared__ float smax[32];
    __shared__ float ssum[32];

    const int tid  = threadIdx.x;
    const int lane = tid & 31;
    const int wave = tid >> 5;
    const int nw   = blockDim.x >> 5;

    float m = -3.402823466e38f;
    for (int j = tid; j < N; j += blockDim.x) m = fmaxf(m, p[j]);
    for (int o = 16; o > 0; o >>= 1) m = fmaxf(m, __shfl_xor(m, o, 32));
    if (lane == 0) smax[wave] = m;
    __syncthreads();
    if (tid < 32) {
        float v = (tid < nw) ? smax[tid] : -3.402823466e38f;
        for (int o = 16; o > 0; o >>= 1) v = fmaxf(v, __shfl_xor(v, o, 32));
        smax[tid] = v;
    }
    __syncthreads();
    m = smax[0];

    float s = 0.f;
    for (int j = tid; j < N; j += blockDim.x) s += expf(p[j] - m);
    for (int o = 16; o > 0; o >>= 1) s += __shfl_xor(s, o, 32);
    if (lane == 0) ssum[wave] = s;
    __syncthreads();
    if (tid < 32) {
        float v = (tid < nw) ? ssum[tid] : 0.f;
        for (int o = 16; o > 0; o >>= 1) v += __shfl_xor(v, o, 32);
        ssum[tid] = v;
    }
    __syncthreads();
    const float lse = m + logf(ssum[0]);

    for (int j = tid; j < N; j += blockDim.x) p[j] = p[j] - lse;
}

// ---------------------------------------------------------------------------
// Host side.  Inputs: x, Wx, bx, Wh, bh, Wc, bc, Wout, bout (all fp32).
// Only batch element 63 contributes to the output (hidden_seq[-1]).
// Workspace: gx (2048x1024 f32 = 8MB) | hidden (2048x256 f32 = 2MB)
// ---------------------------------------------------------------------------
extern "C" void kernel_launch(void* const* d_in, const int* in_sizes, int n_in,
                              void* d_out, int out_size, void* d_ws, size_t ws_size,
                              hipStream_t stream)
{
    constexpr int T = 2048, I = 256, H = 256, O = 512;
    (void)in_sizes; (void)n_in; (void)out_size; (void)ws_size;

    const float* x    = (const float*)d_in[0];
    const float* Wx   = (const float*)d_in[1];
    const float* bx   = (const float*)d_in[2];
    const float* Wh   = (const float*)d_in[3];
    const float* bh   = (const float*)d_in[4];
    const float* Wc   = (const float*)d_in[5];
    const float* bc   = (const float*)d_in[6];
    const float* Wout = (const float*)d_in[7];
    const float* bout = (const float*)d_in[8];

    float* gx     = (float*)d_ws;                                  // T x 4H
    float* hidden = (float*)((char*)d_ws + (size_t)T * 4 * H * 4); // T x H
    float* logits = (float*)d_out;                                 // T x O

    const float* x63 = x + (size_t)63 * T * I;

    // 1) gx = x[63] @ Wx^T + bx   (M=2048, N=1024, K=256)
    {
        int waves  = (T / 16) * ((4 * H) / 64);   // 2048 waves
        int blocks = waves / 8;                   // 8 waves / 256-thread block
        wmma_gemm_f32<256, 1024><<<blocks, 256, 0, stream>>>(x63, Wx, bx, gx, T);
    }

    // 2) sequential scan (one WGP, h/c in LDS)
    lstm_scan<<<1, 1024, 0, stream>>>(gx, Wh, bh, Wc, bc, hidden, T);

    // 3) logits = hidden @ Wout^T + bout   (M=2048, N=512, K=256)
    {
        int waves  = (T / 16) * (O / 64);         // 1024 waves
        int blocks = waves / 8;
        wmma_gemm_f32<256, 512><<<blocks, 256, 0, stream>>>(hidden, Wout, bout, logits, T);
    }

    // 4) in-place log_softmax over each row of 512
    logsoftmax_rows<<<T, 256, 0, stream>>>(logits, O);
}